// PixelContrastLoss1_49503793054185
// MI455X (gfx1250) — compile-verified
//
#include <hip/hip_runtime.h>
#include <math.h>

// ---------------------------------------------------------------------------
// PixelContrastLoss on MI455X (gfx1250, wave32).
// Heavy path: per-class [100x64]x[64x1000] f32 GEMMs via V_WMMA_F32_16X16X4_F32.
// ---------------------------------------------------------------------------

typedef __attribute__((ext_vector_type(2))) float v2f;
typedef __attribute__((ext_vector_type(8))) float v8f;

#define NVOX   (96 * 96 * 96)   // 884736 voxels per batch
#define NCH    64
#define NBATCH 2
#define NCLS   7
#define NHARD  25
#define NEASY  25
#define MROWS  100              // B * NV anchors per class
#define MPAD   112              // 7 tiles of 16
#define MTILES 7
#define QHALF  1000
#define NPAD   1008             // 63 tiles of 16
#define NTILES 63
#define INV_TEMP 10.0f

// workspace layout (bytes, 256-aligned regions except OFF_PART)
static constexpr size_t OFF_IDX  = 0;                     // int[2][7][50]          = 2800 B
static constexpr size_t OFF_A    = 2816;                  // f32[7][112][64]        = 200704 B
static constexpr size_t OFF_P    = OFF_A  + 200704;       // f32[7][1008][64]       = 1806336 B
static constexpr size_t OFF_NG   = OFF_P  + 1806336;      // f32[7][1008][64]
static constexpr size_t OFF_NEG  = OFF_NG + 1806336;      // f32[7][112]            = 3136 B
static constexpr size_t OFF_PART = OFF_NEG + 3136;        // f32[49]
// total ~3.82 MB

// ---------------------------------------------------------------------------
// K1: ordered "first 25 hard + first 25 easy" index selection per (b, class).
// One 256-thread block (8 wave32s) per (b,c). Stable prefix over index order,
// early exit once both pools are full. Deterministic.
// ---------------------------------------------------------------------------
__global__ void k_select(const int* __restrict__ labels,
                         const int* __restrict__ predict,
                         int* __restrict__ idxbuf) {
  const int bc   = blockIdx.x;          // 0..13
  const int b    = bc / NCLS;
  const int cls  = bc % NCLS;           // 0..6
  const int cval = cls + 1;             // class label 1..7
  const int tid  = threadIdx.x;
  const int wave = tid >> 5;
  const int lane = tid & 31;

  const int* l = labels  + (size_t)b * NVOX;
  const int* p = predict + (size_t)b * NVOX;
  int* outH = idxbuf + (b * NCLS + cls) * (NHARD + NEASY);
  int* outE = outH + NHARD;

  __shared__ int sHard, sEasy;
  __shared__ int wTotH[8], wTotE[8];

  if (tid == 0) { sHard = 0; sEasy = 0; }
  if (tid < NHARD + NEASY) outH[tid] = 0;   // fallback init (never needed statistically)
  __syncthreads();

  for (int base = 0; base < NVOX; base += 256) {
    const int pos = base + tid;
    bool mH = false, mE = false;
    if (pos < NVOX) {
      const int lv = l[pos];
      const int pv = p[pos];
      mH = (lv == cval) && (pv != cval);
      mE = (lv == cval) && (pv == cval);
    }
    const unsigned bH = (unsigned)__ballot(mH);
    const unsigned bE = (unsigned)__ballot(mE);
    const unsigned mlt = (1u << lane) - 1u;
    const int preH = __popc(bH & mlt);
    const int preE = __popc(bE & mlt);
    if (lane == 0) { wTotH[wave] = __popc(bH); wTotE[wave] = __popc(bE); }
    __syncthreads();

    int offH = sHard, offE = sEasy;
    for (int w = 0; w < wave; ++w) { offH += wTotH[w]; offE += wTotE[w]; }
    int totH = 0, totE = 0;
    for (int w = 0; w < 8; ++w) { totH += wTotH[w]; totE += wTotE[w]; }

    if (mH) { const int s = offH + preH; if (s < NHARD) outH[s] = pos; }
    if (mE) { const int s = offE + preE; if (s < NEASY) outE[s] = pos; }
    __syncthreads();
    if (tid == 0) { sHard += totH; sEasy += totE; }
    __syncthreads();
    if (sHard >= NHARD && sEasy >= NEASY) break;
  }
}

// ---------------------------------------------------------------------------
// K2: gather anchor vectors from feats [B,C,S,S,S], L2-normalize, write
// A[7][112][64]; padded rows (100..111) are zero. One wave per row.
// ---------------------------------------------------------------------------
__global__ void k_anchors(const float* __restrict__ feats,
                          const int* __restrict__ idxbuf,
                          float* __restrict__ A) {
  const int row  = blockIdx.x * 8 + (threadIdx.x >> 5);   // 0..783
  const int lane = threadIdx.x & 31;
  const int c = row / MPAD;
  const int j = row % MPAD;

  float x0 = 0.f, x1 = 0.f;
  if (j < MROWS) {
    const int b   = j / 50;   // NV = 50
    const int v   = j % 50;
    const int idx = idxbuf[(b * NCLS + c) * 50 + v];
    const float* fb = feats + ((size_t)b * NCH) * (size_t)NVOX + (size_t)idx;
    x0 = fb[(size_t)(2 * lane) * NVOX];
    x1 = fb[(size_t)(2 * lane + 1) * NVOX];
    float ss = x0 * x0 + x1 * x1;
    for (int m = 1; m < 32; m <<= 1) ss += __shfl_xor(ss, m);
    const float inv = 1.0f / fmaxf(sqrtf(ss), 1e-12f);
    x0 *= inv; x1 *= inv;
  }
  float2* dst = (float2*)(A + (size_t)row * NCH);
  dst[lane] = make_float2(x0, x1);
}

// ---------------------------------------------------------------------------
// K3: L2-normalize queue rows into padded P[7][1008][64] / Ng[7][1008][64].
// One wave per destination row; padded rows zero.
// ---------------------------------------------------------------------------
__global__ void k_queue(const float* __restrict__ queue,
                        float* __restrict__ P,
                        float* __restrict__ Ng) {
  const int row  = blockIdx.x * 8 + (threadIdx.x >> 5);   // 0..14111
  const int lane = threadIdx.x & 31;
  const int c = row / (2 * NPAD);
  const int t = row % (2 * NPAD);

  float* dst;
  const float* src = nullptr;
  if (t < NPAD) {
    dst = P + ((size_t)c * NPAD + t) * NCH;
    if (t < QHALF) src = queue + ((size_t)(1 + c) * 2000 + t) * NCH;
  } else {
    const int u = t - NPAD;
    dst = Ng + ((size_t)c * NPAD + u) * NCH;
    if (u < QHALF) src = queue + ((size_t)(1 + c) * 2000 + QHALF + u) * NCH;
  }

  float2 v = make_float2(0.f, 0.f);
  if (src) {
    v = ((const float2*)src)[lane];
    float ss = v.x * v.x + v.y * v.y;
    for (int m = 1; m < 32; m <<= 1) ss += __shfl_xor(ss, m);
    const float inv = 1.0f / fmaxf(sqrtf(ss), 1e-12f);
    v.x *= inv; v.y *= inv;
  }
  ((float2*)dst)[lane] = v;
}

// ---------------------------------------------------------------------------
// K4/K5: per-class GEMM via V_WMMA_F32_16X16X4_F32 (K=64 -> 16 chained WMMAs).
// Block = (class, M-tile); 8 waves stride the 63 N-tiles.
//   mode 0: negsum[c][m] = sum_n exp(logit)            (B = normalized negatives)
//   mode 1: partials[blk] = sum(-logit + log(exp(logit)+negsum))  (B = positives)
// A frag per lane: row = mtile*16 + (lane&15), K pair at 4k + (lane>=16 ? 2:0).
// B frag per lane: col = ntile*16 + (lane&15), same K-pair pattern (B is K x N
// with B[k][n] = P[n][k], so the row-major [n][64] buffer loads symmetrically).
// D layout: vgpr v, lane L -> row v + 8*(L/16), col L%16.
// ---------------------------------------------------------------------------
__global__ void k_gemm(const float* __restrict__ A,
                       const float* __restrict__ Bmat,
                       float* __restrict__ negsum,
                       float* __restrict__ partials,
                       int mode) {
  const int blk   = blockIdx.x;        // 0..48
  const int c     = blk / MTILES;
  const int mtile = blk % MTILES;
  const int wave  = threadIdx.x >> 5;
  const int lane  = threadIdx.x & 31;
  const int halfhi = (lane >= 16) ? 1 : 0;
  const int rbase  = halfhi * 8;

  const float* Ac = A    + (size_t)c * MPAD * NCH;
  const float* Bc = Bmat + (size_t)c * NPAD * NCH;

  __shared__ float sneg[16];
  __shared__ float sred[8][16];
  __shared__ float swv[8];

  if (mode == 1 && threadIdx.x < 16)
    sneg[threadIdx.x] = negsum[c * MPAD + mtile * 16 + threadIdx.x];
  __syncthreads();

  // preload 16 A fragments for this M-tile
  const int arow = mtile * 16 + (lane & 15);
  const float* ap = Ac + (size_t)arow * NCH + 2 * halfhi;
  v2f afrag[16];
#pragma unroll
  for (int k = 0; k < 16; ++k) afrag[k] = *(const v2f*)(ap + 4 * k);

  float acc[8];
#pragma unroll
  for (int v = 0; v < 8; ++v) acc[v] = 0.f;

  for (int nt = wave; nt < NTILES; nt += 8) {
    const int col = nt * 16 + (lane & 15);
    const float* bp = Bc + (size_t)col * NCH + 2 * halfhi;
    v8f d = {};
#pragma unroll
    for (int k = 0; k < 16; ++k) {
      const v2f bfrag = *(const v2f*)(bp + 4 * k);
      d = __builtin_amdgcn_wmma_f32_16x16x4_f32(
              false, afrag[k], false, bfrag, (short)0, d, false, false);
    }
    const bool colValid = (nt * 16 + (lane & 15)) < QHALF;
#pragma unroll
    for (int v = 0; v < 8; ++v) {
      const float logit = d[v] * INV_TEMP;
      if (mode == 0) {
        if (colValid) acc[v] += expf(logit);
      } else {
        const int grow = mtile * 16 + rbase + v;
        if (colValid && grow < MROWS) {
          const float ns = sneg[rbase + v];
          acc[v] += -logit + logf(expf(logit) + ns);
        }
      }
    }
  }

  if (mode == 0) {
    // per-row sums: reduce over the 16 lanes of each half-wave
#pragma unroll
    for (int v = 0; v < 8; ++v) {
      float s = acc[v];
      for (int m = 1; m < 16; m <<= 1) s += __shfl_xor(s, m);
      acc[v] = s;
    }
    if ((lane & 15) == 0)
      for (int v = 0; v < 8; ++v) sred[wave][rbase + v] = acc[v];
    __syncthreads();
    if (threadIdx.x < 16) {
      float s = 0.f;
      for (int w = 0; w < 8; ++w) s += sred[w][threadIdx.x];
      negsum[c * MPAD + mtile * 16 + threadIdx.x] = s;
    }
  } else {
    float s = 0.f;
#pragma unroll
    for (int v = 0; v < 8; ++v) s += acc[v];
    for (int m = 1; m < 32; m <<= 1) s += __shfl_xor(s, m);
    if (lane == 0) swv[wave] = s;
    __syncthreads();
    if (threadIdx.x == 0) {
      float t = 0.f;
      for (int w = 0; w < 8; ++w) t += swv[w];
      partials[blk] = t;
    }
  }
}

// ---------------------------------------------------------------------------
// K6: fixed-order final reduction -> mean.
// ---------------------------------------------------------------------------
__global__ void k_final(const float* __restrict__ partials, float* __restrict__ out) {
  if (threadIdx.x == 0) {
    float s = 0.f;
    for (int i = 0; i < NCLS * MTILES; ++i) s += partials[i];
    out[0] = s / (float)(NCLS * MROWS * QHALF);
  }
}

// ---------------------------------------------------------------------------
extern "C" void kernel_launch(void* const* d_in, const int* in_sizes, int n_in,
                              void* d_out, int out_size, void* d_ws, size_t ws_size,
                              hipStream_t stream) {
  (void)in_sizes; (void)n_in; (void)out_size; (void)ws_size;
  const float* feats   = (const float*)d_in[0];   // [2,64,96,96,96]
  const int*   labels  = (const int*)d_in[1];     // [2,96,96,96]
  const int*   predict = (const int*)d_in[2];     // [2,96,96,96]
  const float* queue   = (const float*)d_in[3];   // [8,2000,64]

  char* ws = (char*)d_ws;
  int*   idxbuf   = (int*)(ws + OFF_IDX);
  float* A        = (float*)(ws + OFF_A);
  float* P        = (float*)(ws + OFF_P);
  float* Ng       = (float*)(ws + OFF_NG);
  float* negsum   = (float*)(ws + OFF_NEG);
  float* partials = (float*)(ws + OFF_PART);
  float* out      = (float*)d_out;

  k_select <<<NBATCH * NCLS,            256, 0, stream>>>(labels, predict, idxbuf);
  k_anchors<<<(NCLS * MPAD) / 8,        256, 0, stream>>>(feats, idxbuf, A);
  k_queue  <<<(NCLS * 2 * NPAD) / 8,    256, 0, stream>>>(queue, P, Ng);
  k_gemm   <<<NCLS * MTILES,            256, 0, stream>>>(A, Ng, negsum, partials, 0);
  k_gemm   <<<NCLS * MTILES,            256, 0, stream>>>(A, P,  negsum, partials, 1);
  k_final  <<<1, 32, 0, stream>>>(partials, out);
}